// IPECosTheta_12309376270483
// MI455X (gfx1250) — compile-verified
//
#include <hip/hip_runtime.h>
#include <math.h>

// out[e] = log( sum_k x^k * t_k(b) ),  t_k(b) = sum_m K[m,k] b^m,
// K[m,k] = W[m] * R[m,k]  (exact Pochhammer recurrences, no lgamma needed):
//   W[0] = Gamma(0.5)Gamma(4.5)/Gamma(5) = (105/384)*pi
//   W[m+1] = W[m] * (2m+0.5)(2m+1.5) / (4*(2m+5)(2m+6)*(m+1)^2)
//   R[m,0] = 1;  R[m,k+1] = R[m,k] * (4.5+k) / ((0.5+k)(2m+5+k)(k+1))
// x = a*a/4. Inner contraction is an fp32 GEMM via V_WMMA_F32_16X16X4_F32:
//   T[e,k] = sum_m Bpow[e,m] * K[m,k]   (M=16 elems, N=16 cols, K-dim=4 per step)
//
// Occupancy note: no min-waves launch bound on purpose. Capping VGPRs at
// 256 (min 4 waves) or ~341 (min 3 waves) makes the scheduler spill to
// scratch inside the hot loop (vmem 39-53 vs 7). The unconstrained build
// uses ~430 VGPRs with zero spills; the s_set_vgpr_msb prefixes it needs
// are SALU ops that co-issue for free next to the WMMA/VALU stream.

typedef __attribute__((ext_vector_type(2))) float v2f;
typedef __attribute__((ext_vector_type(8))) float v8f;

#define NTERMS 20
#define KPAD   32   // K matrix columns padded with zeros (N tile 16..31)
#define TROW   36   // padded row stride (floats) for the LDS transpose buffer

__global__ __launch_bounds__(256) void ipe_logsumexp_wmma(
    const float* __restrict__ a, const float* __restrict__ b,
    float* __restrict__ out, int nElem, int nChunks) {
  __shared__ __align__(16) float Ksh[NTERMS][KPAD];       // 2560 B
  __shared__ __align__(16) float Tsh[8][2][16][TROW];     // 36864 B

  // ---- Build constant table once per block: pure f64 mul/div recurrences ----
  if (threadIdx.x < NTERMS) {
    const int m = threadIdx.x;
    double W = 0.2734375 * 3.14159265358979323846;        // W[0]
    for (int j = 0; j < m; ++j) {
      double tj = 2.0 * j;
      W *= (tj + 0.5) * (tj + 1.5) /
           (4.0 * (tj + 5.0) * (tj + 6.0) * (j + 1.0) * (j + 1.0));
    }
    double R = W;
    for (int k = 0; k < NTERMS; ++k) {
      Ksh[m][k] = (float)R;
      R *= (4.5 + k) / ((0.5 + k) * (2.0 * m + 5.0 + k) * (k + 1.0));
    }
    for (int k = NTERMS; k < KPAD; ++k) Ksh[m][k] = 0.0f; // zero pad cols 20..31
  }
  __syncthreads();

  const int lane = threadIdx.x & 31;
  const int wv   = threadIdx.x >> 5;
  const int col  = lane & 15;
  const int rsel = (lane < 16) ? 0 : 2;  // B operand: vgpr0 holds rows {0,2}, vgpr1 rows {1,3}

  // Loop-invariant B operands: WMMA-K dim = m (rows of K), N = k_out (cols)
  v2f B0[5], B1[5];
#pragma unroll
  for (int s = 0; s < 5; ++s) {
    B0[s] = v2f{Ksh[4 * s + rsel][col],      Ksh[4 * s + rsel + 1][col]};
    B1[s] = v2f{Ksh[4 * s + rsel][16 + col], Ksh[4 * s + rsel + 1][16 + col]};
  }

  const int nwaves = (int)gridDim.x * (blockDim.x >> 5);
  const int gwave  = (int)blockIdx.x * (blockDim.x >> 5) + wv;

  for (int chunk = gwave; chunk < nChunks; chunk += nwaves) {
    const int base = chunk * 32;
    if (base + 32 <= nElem) {  // wave-uniform: EXEC all ones for WMMA
      // Two 16-element tiles per wave (A = powers of b, per ISA 16x4 layout)
      float bA = b[base + col];
      float bB = b[base + 16 + col];
      float aX = a[base + lane];
      float x  = aX * aX * 0.25f;

      float bA2 = bA * bA, bB2 = bB * bB;
      float p0A = (lane < 16) ? 1.0f : bA2;        // b^{K-col 0 or 2}
      float p1A = (lane < 16) ? bA   : bA2 * bA;   // b^{K-col 1 or 3}
      float p0B = (lane < 16) ? 1.0f : bB2;
      float p1B = (lane < 16) ? bB   : bB2 * bB;
      float b4A = bA2 * bA2, b4B = bB2 * bB2;

      v8f c0A = {}, c1A = {}, c0B = {}, c1B = {};
#pragma unroll
      for (int s = 0; s < 5; ++s) {  // chain K-dim: m = 4s..4s+3
        v2f Aa = v2f{p0A, p1A};
        v2f Ab = v2f{p0B, p1B};
        c0A = __builtin_amdgcn_wmma_f32_16x16x4_f32(false, Aa, false, B0[s], (short)0, c0A, false, false);
        c1A = __builtin_amdgcn_wmma_f32_16x16x4_f32(false, Aa, false, B1[s], (short)0, c1A, false, false);
        c0B = __builtin_amdgcn_wmma_f32_16x16x4_f32(false, Ab, false, B0[s], (short)0, c0B, false, false);
        c1B = __builtin_amdgcn_wmma_f32_16x16x4_f32(false, Ab, false, B1[s], (short)0, c1B, false, false);
        p0A *= b4A; p1A *= b4A;
        p0B *= b4B; p1B *= b4B;
      }

      // Transpose D (elem-rows striped over lanes) through wave-private LDS slice.
      // LDS ops are in-order within a wave; no barrier needed.
      const int row = (lane < 16) ? 0 : 8;  // C/D layout: vgpr i = rows {i, i+8}
#pragma unroll
      for (int i = 0; i < 8; ++i) {
        Tsh[wv][0][row + i][col]      = c0A[i];
        Tsh[wv][0][row + i][16 + col] = c1A[i];
        Tsh[wv][1][row + i][col]      = c0B[i];
        Tsh[wv][1][row + i][16 + col] = c1B[i];
      }

      // Each lane finalizes its own element: T[e,0..19] via ds_load_b128 x5,
      // fully unrolled Horner in x, then log.
      const float4* pT4 = (const float4*)&Tsh[wv][lane >> 4][lane & 15][0];
      float4 q0 = pT4[0], q1 = pT4[1], q2 = pT4[2], q3 = pT4[3], q4 = pT4[4];
      float v = q4.w;
      v = fmaf(v, x, q4.z); v = fmaf(v, x, q4.y); v = fmaf(v, x, q4.x);
      v = fmaf(v, x, q3.w); v = fmaf(v, x, q3.z); v = fmaf(v, x, q3.y); v = fmaf(v, x, q3.x);
      v = fmaf(v, x, q2.w); v = fmaf(v, x, q2.z); v = fmaf(v, x, q2.y); v = fmaf(v, x, q2.x);
      v = fmaf(v, x, q1.w); v = fmaf(v, x, q1.z); v = fmaf(v, x, q1.y); v = fmaf(v, x, q1.x);
      v = fmaf(v, x, q0.w); v = fmaf(v, x, q0.z); v = fmaf(v, x, q0.y); v = fmaf(v, x, q0.x);
      out[base + lane] = logf(v);
    } else {
      // Tail: per-lane scalar path (same math, pure VALU)
      int e = base + lane;
      if (e < nElem) {
        float bb = b[e], aa = a[e];
        float x = aa * aa * 0.25f;
        float accv = 0.0f, xp = 1.0f;
        for (int k = 0; k < NTERMS; ++k) {
          float t = 0.0f, bp = 1.0f;
          for (int m = 0; m < NTERMS; ++m) { t = fmaf(Ksh[m][k], bp, t); bp *= bb; }
          accv = fmaf(t, xp, accv);
          xp *= x;
        }
        out[e] = logf(accv);
      }
    }
  }
}

extern "C" void kernel_launch(void* const* d_in, const int* in_sizes, int n_in,
                              void* d_out, int out_size, void* d_ws, size_t ws_size,
                              hipStream_t stream) {
  (void)n_in; (void)out_size; (void)d_ws; (void)ws_size;
  const float* a = (const float*)d_in[0];
  const float* b = (const float*)d_in[1];
  float* out = (float*)d_out;
  int n = in_sizes[0];
  int chunks = (n + 31) / 32;              // 32 elements per wave-iteration
  int blocks = (chunks + 7) / 8;           // 8 waves per 256-thread block
  if (blocks > 1024) blocks = 1024;        // grid-stride beyond this
  if (blocks < 1) blocks = 1;
  ipe_logsumexp_wmma<<<blocks, 256, 0, stream>>>(a, b, out, n, chunks);
}